// MyNet_36386962932140
// MI455X (gfx1250) — compile-verified
//
#include <hip/hip_runtime.h>
#include <math.h>

typedef __attribute__((ext_vector_type(2))) float v2f;
typedef __attribute__((ext_vector_type(8))) float v8f;

// ---------------------------------------------------------------------------
// Degree + normalization
// ---------------------------------------------------------------------------
__global__ void degree_kernel(const int* __restrict__ dst, float* __restrict__ deg, int E) {
    int i = blockIdx.x * blockDim.x + threadIdx.x;
    if (i < E) atomicAdd(&deg[dst[i]], 1.0f);
}

__global__ void dis_kernel(float* __restrict__ deg, int N) {
    int i = blockIdx.x * blockDim.x + threadIdx.x;
    if (i < N) deg[i] = rsqrtf(deg[i] + 1.0f);
}

// ---------------------------------------------------------------------------
// FP32 WMMA GEMM:  H[M x N] = X[M x 48] * W[48 x N]
// One wave computes a 16-row strip; N/16 accumulator tiles; K stepped by 4.
// Fragment layouts per CDNA5 ISA 7.12.2 (32-bit A 16x4, 32-bit C/D 16x16).
// ---------------------------------------------------------------------------
template <int N>
__global__ void gemm_wmma_f32(const float* __restrict__ X, const float* __restrict__ W,
                              float* __restrict__ H, int M) {
    constexpr int K = 48;
    constexpr int NT = N / 16;

    const int wave = (int)((blockIdx.x * blockDim.x + threadIdx.x) >> 5);
    const int lane = threadIdx.x & 31;
    const int m0 = wave * 16;
    if (m0 >= M) return;                 // wave-uniform exit: EXEC all-ones for WMMA

    const int half = lane >> 4;          // 0: K=0,1  1: K=2,3 within each K-step
    const int l16  = lane & 15;
    const int row  = m0 + l16;           // A-matrix row owned by this lane
    const int koff = half * 2;

    v8f acc[NT];
    const v8f vzero = {0.f, 0.f, 0.f, 0.f, 0.f, 0.f, 0.f, 0.f};
#pragma unroll
    for (int t = 0; t < NT; ++t) acc[t] = vzero;

#pragma unroll
    for (int k = 0; k < K; k += 4) {
        // A fragment: lane<16 -> A[row][k..k+1], lane>=16 -> A[row][k+2..k+3]
        v2f a;
        a.x = X[row * K + k + koff];
        a.y = X[row * K + k + koff + 1];
#pragma unroll
        for (int t = 0; t < NT; ++t) {
            const int col = t * 16 + l16;
            // B fragment (4x16, KxN): mirrors A striping over lane halves
            v2f b;
            b.x = W[(k + koff) * N + col];
            b.y = W[(k + koff + 1) * N + col];
            acc[t] = __builtin_amdgcn_wmma_f32_16x16x4_f32(
                false, a, false, b, (short)0, acc[t], false, false);
        }
    }

    // C/D layout: VGPR v -> row m0+v (lanes 0-15) or m0+v+8 (lanes 16-31)
    const int rbase = m0 + half * 8;
#pragma unroll
    for (int t = 0; t < NT; ++t) {
#pragma unroll
        for (int v = 0; v < 8; ++v) {
            H[(rbase + v) * N + t * 16 + l16] = acc[t][v];
        }
    }
}

// ---------------------------------------------------------------------------
// Edge scatter: agg[dst] += dis[src]*dis[dst] * h[src]   (per edge,feature)
// Consecutive lanes share an edge -> coalesced gather + clustered atomics.
// ---------------------------------------------------------------------------
template <int F>
__global__ void scatter_kernel(const float* __restrict__ h,
                               const int* __restrict__ src,
                               const int* __restrict__ dst,
                               const float* __restrict__ dis,
                               float* __restrict__ agg, int E) {
    unsigned idx = blockIdx.x * blockDim.x + threadIdx.x;
    unsigned total = (unsigned)E * (unsigned)F;
    if (idx >= total) return;
    int e = idx / F;
    int k = idx % F;
    int s = src[e];
    int d = dst[e];
    float norm = dis[s] * dis[d];
    atomicAdd(&agg[d * F + k], h[s * F + k] * norm);
}

// ---------------------------------------------------------------------------
// Finalize: out = [relu]( agg + dis^2 * h + b )
// ---------------------------------------------------------------------------
template <int F, bool RELU>
__global__ void finalize_kernel(const float* __restrict__ agg,
                                const float* __restrict__ h,
                                const float* __restrict__ dis,
                                const float* __restrict__ bias,
                                float* __restrict__ out, int Nn) {
    int idx = blockIdx.x * blockDim.x + threadIdx.x;
    if (idx >= Nn * F) return;
    int i = idx / F;
    int k = idx % F;
    float di = dis[i];
    float v = agg[idx] + h[idx] * di * di + bias[k];
    out[idx] = RELU ? fmaxf(v, 0.0f) : v;
}

// ---------------------------------------------------------------------------
// In-place log-softmax over 16 features, one thread per node.
// ---------------------------------------------------------------------------
__global__ void logsoftmax16_kernel(float* __restrict__ x, int Nn) {
    int i = blockIdx.x * blockDim.x + threadIdx.x;
    if (i >= Nn) return;
    float4* p = (float4*)(x + i * 16);
    float4 r[4];
#pragma unroll
    for (int q = 0; q < 4; ++q) r[q] = p[q];

    float m = r[0].x;
#pragma unroll
    for (int q = 0; q < 4; ++q) {
        m = fmaxf(m, r[q].x); m = fmaxf(m, r[q].y);
        m = fmaxf(m, r[q].z); m = fmaxf(m, r[q].w);
    }
    float s = 0.f;
#pragma unroll
    for (int q = 0; q < 4; ++q) {
        s += __expf(r[q].x - m) + __expf(r[q].y - m) +
             __expf(r[q].z - m) + __expf(r[q].w - m);
    }
    float lse = m + logf(s);
#pragma unroll
    for (int q = 0; q < 4; ++q) {
        r[q].x -= lse; r[q].y -= lse; r[q].z -= lse; r[q].w -= lse;
        p[q] = r[q];
    }
}

// ---------------------------------------------------------------------------
// Driver
// ---------------------------------------------------------------------------
static inline int cdiv(long long a, int b) { return (int)((a + b - 1) / b); }

extern "C" void kernel_launch(void* const* d_in, const int* in_sizes, int n_in,
                              void* d_out, int out_size, void* d_ws, size_t ws_size,
                              hipStream_t stream) {
    const float* x   = (const float*)d_in[0];
    const int*   ei  = (const int*)d_in[1];
    const float* W1  = (const float*)d_in[2];
    const float* b1  = (const float*)d_in[3];
    const float* W2  = (const float*)d_in[4];
    const float* b2  = (const float*)d_in[5];
    const float* W3  = (const float*)d_in[6];
    const float* b3  = (const float*)d_in[7];
    float* out = (float*)d_out;

    const int N = in_sizes[0] / 48;      // 50000 nodes
    const int E = in_sizes[1] / 2;       // 1.6M edges
    const int* src = ei;
    const int* dst = ei + E;

    // Workspace layout (floats)
    float* ws  = (float*)d_ws;
    float* dis = ws;                     // N       (degree, then rsqrt)
    float* h   = dis + N;                // N*48
    float* agg = h + (size_t)N * 48;     // N*48
    float* x1  = agg + (size_t)N * 48;   // N*48
    float* x2  = x1 + (size_t)N * 48;    // N*48

    const int B = 256;

    // --- normalization ---
    hipMemsetAsync(dis, 0, (size_t)N * sizeof(float), stream);
    degree_kernel<<<cdiv(E, B), B, 0, stream>>>(dst, dis, E);
    dis_kernel<<<cdiv(N, B), B, 0, stream>>>(dis, N);

    const int gemm_blocks = cdiv((long long)(N / 16) * 32, B);

    // --- layer 1: 48 -> 48, ReLU ---
    gemm_wmma_f32<48><<<gemm_blocks, B, 0, stream>>>(x, W1, h, N);
    hipMemsetAsync(agg, 0, (size_t)N * 48 * sizeof(float), stream);
    scatter_kernel<48><<<cdiv((long long)E * 48, B), B, 0, stream>>>(h, src, dst, dis, agg, E);
    finalize_kernel<48, true><<<cdiv((long long)N * 48, B), B, 0, stream>>>(agg, h, dis, b1, x1, N);

    // --- layer 2: 48 -> 48, ReLU ---
    gemm_wmma_f32<48><<<gemm_blocks, B, 0, stream>>>(x1, W2, h, N);
    hipMemsetAsync(agg, 0, (size_t)N * 48 * sizeof(float), stream);
    scatter_kernel<48><<<cdiv((long long)E * 48, B), B, 0, stream>>>(h, src, dst, dis, agg, E);
    finalize_kernel<48, true><<<cdiv((long long)N * 48, B), B, 0, stream>>>(agg, h, dis, b2, x2, N);

    // --- layer 3: 48 -> 16, no ReLU, then log-softmax ---
    gemm_wmma_f32<16><<<gemm_blocks, B, 0, stream>>>(x2, W3, h, N);
    hipMemsetAsync(agg, 0, (size_t)N * 16 * sizeof(float), stream);
    scatter_kernel<16><<<cdiv((long long)E * 16, B), B, 0, stream>>>(h, src, dst, dis, agg, E);
    finalize_kernel<16, false><<<cdiv((long long)N * 16, B), B, 0, stream>>>(agg, h, dis, b3, out, N);

    logsoftmax16_kernel<<<cdiv(N, B), B, 0, stream>>>(out, N);
}